// BarrelShifterRight64_23407571764143
// MI455X (gfx1250) — compile-verified
//
#include <hip/hip_runtime.h>

// 64-wide barrel shifter over 0/1-valued float rows.
//   out[row][k] = (k >= s) ? data[row][k-s] : 0 ,  s = 6-bit MSB-first control.
//
// Formulation: a 0/1 float row IS a 64-bit word. Pack with two wave32 ballots
// (bit k = element k), barrel-shift with ONE scalar s_lshl_b64 (zero-fills the
// low end exactly like the reference), unpack 2 bits/lane, store aligned b64.
//
// Efficiency round:
//  - 8 rows per wave; two ballots decode all eight 6-bit shift amounts
//    (control lanes mapped r=lane>>3, j=lane&7 -> no integer division;
//    each group loads its control word reversed so s = (bal >> 8r) & 63).
//  - readfirstlane(wave id) forces row0/base pointers into SGPRs: loads and
//    stores become saddr-form with immediate offsets, address math on SALU.
//  - main kernel handles only FULL 64-row blocks (no tail logic in the hot
//    path); a per-row tail kernel is launched only when needed (never for
//    the reference B = 2,097,152).
// HBM-bound: ~1.12 GB traffic -> ~48us floor at 23.3 TB/s; NT hints keep the
// 512MB streams out of the 192MB L2.

#define NBITS 64
#define NCTRL 6
#define WAVES_PER_BLOCK 8
#define ROWS_PER_WAVE 8
#define ROWS_PER_BLOCK (WAVES_PER_BLOCK * ROWS_PER_WAVE)   // 64

typedef float v2f __attribute__((ext_vector_type(2)));

__global__ __launch_bounds__(256) void barrel64_main_kernel(
    const float* __restrict__ data,
    const float* __restrict__ shift,
    float* __restrict__ out)
{
    const int tid  = threadIdx.x;
    const int lane = tid & 31;
    // wave index is uniform; tell the compiler so bases land in SGPRs
    const int wv = __builtin_amdgcn_readfirstlane(tid >> 5);
    const long long row0 =
        ((long long)blockIdx.x * WAVES_PER_BLOCK + wv) * ROWS_PER_WAVE;

    // ---- eight control words via two loads + two ballots.
    //      group r = lane>>3 (rows r and r+4), slot j = lane&7, active j<6.
    //      lane tests shift[row*6 + (5-j)] so ballot bit (8r+j) has weight
    //      1<<j within that row's control word.
    const int r = lane >> 3;
    const int j = lane & 7;
    const float* sb = shift + row0 * NCTRL;
    float sv0 = 0.0f, sv1 = 0.0f;
    if (j < NCTRL) {
        const int o = r * NCTRL + (NCTRL - 1 - j);
        sv0 = __builtin_nontemporal_load(sb + o);                      // rows 0..3
        sv1 = __builtin_nontemporal_load(sb + 4 * NCTRL + o);          // rows 4..7
    }
    const unsigned bal0 = __builtin_amdgcn_ballot_w32(sv0 > 0.5f);
    const unsigned bal1 = __builtin_amdgcn_ballot_w32(sv1 > 0.5f);

    const float* dbase = data + row0 * NBITS;
    float*       obase = out  + row0 * NBITS;
    const int twoL = lane * 2;

#pragma unroll
    for (int rr = 0; rr < ROWS_PER_WAVE; ++rr) {
        // pack row rr: lane L supplies elements L and L+32
        const float d0 = __builtin_nontemporal_load(dbase + rr * NBITS + lane);
        const float d1 = __builtin_nontemporal_load(dbase + rr * NBITS + lane + 32);
        const unsigned b0 = __builtin_amdgcn_ballot_w32(d0 > 0.5f);
        const unsigned b1 = __builtin_amdgcn_ballot_w32(d1 > 0.5f);

        const unsigned s =
            (((rr < 4) ? bal0 : bal1) >> (8 * (rr & 3))) & 63u;
        const unsigned long long m =
            (((unsigned long long)b1 << 32) | (unsigned long long)b0) << s;

        // unpack bits {2L, 2L+1}; u32->f32 of 0/1 is exact
        const unsigned t = (unsigned)(m >> twoL);
        v2f res;
        res.x = (float)(t & 1u);
        res.y = (float)((t >> 1) & 1u);
        __builtin_nontemporal_store(res, (v2f*)(obase + rr * NBITS + twoL));
    }
}

// Per-row tail kernel (only launched when nrows % 64 != 0).
__global__ __launch_bounds__(256) void barrel64_tail_kernel(
    const float* __restrict__ data,
    const float* __restrict__ shift,
    float* __restrict__ out,
    long long row_start,
    long long nrows)
{
    const int lane = threadIdx.x & 31;
    const long long row =
        row_start + (long long)blockIdx.x * WAVES_PER_BLOCK + (threadIdx.x >> 5);
    if (row >= nrows) return;  // wave-uniform

    float sv = 0.0f;
    if (lane < NCTRL)
        sv = __builtin_nontemporal_load(&shift[row * NCTRL + (NCTRL - 1 - lane)]);
    const unsigned s = __builtin_amdgcn_ballot_w32(sv > 0.5f) & 63u;

    const float d0 = __builtin_nontemporal_load(&data[row * NBITS + lane]);
    const float d1 = __builtin_nontemporal_load(&data[row * NBITS + 32 + lane]);
    const unsigned b0 = __builtin_amdgcn_ballot_w32(d0 > 0.5f);
    const unsigned b1 = __builtin_amdgcn_ballot_w32(d1 > 0.5f);
    const unsigned long long m =
        (((unsigned long long)b1 << 32) | (unsigned long long)b0) << s;

    const unsigned t = (unsigned)(m >> (2 * lane));
    v2f res;
    res.x = (float)(t & 1u);
    res.y = (float)((t >> 1) & 1u);
    __builtin_nontemporal_store(res, (v2f*)(out + row * NBITS + 2 * lane));
}

extern "C" void kernel_launch(void* const* d_in, const int* in_sizes, int n_in,
                              void* d_out, int out_size, void* d_ws, size_t ws_size,
                              hipStream_t stream) {
    const float* data  = (const float*)d_in[0];   // (B, 64) float32 of 0/1
    const float* shift = (const float*)d_in[1];   // (B, 6)  float32 of 0/1
    float* out = (float*)d_out;                   // (B, 64) float32

    const long long nrows  = (long long)in_sizes[0] / NBITS;
    const long long nfull  = nrows / ROWS_PER_BLOCK;          // full 64-row blocks
    const long long done   = nfull * ROWS_PER_BLOCK;
    const long long tail   = nrows - done;

    if (nfull > 0) {
        barrel64_main_kernel<<<dim3((unsigned)nfull), dim3(256), 0, stream>>>(
            data, shift, out);
    }
    if (tail > 0) {
        const long long tblocks = (tail + WAVES_PER_BLOCK - 1) / WAVES_PER_BLOCK;
        barrel64_tail_kernel<<<dim3((unsigned)tblocks), dim3(256), 0, stream>>>(
            data, shift, out, done, nrows);
    }
}